// GATv2_89704686944361
// MI455X (gfx1250) — compile-verified
//
#include <hip/hip_runtime.h>
#include <math.h>

// ---------------------------------------------------------------------------
// GATv2 (5 layers) for gfx1250 / MI455X.
//   N=50000 nodes, E=400000 edges (+N self loops), F=128, H=8, C=8, D=64, NC=40
// GEMMs: V_WMMA_F32_16X16X4_F32 (exact fp32), W slab staged in LDS.
// Edge softmax: 2 passes (logits+atomicMax, fused exp+denom+scatter), division
// deferred to finalize. All hot per-edge state (xl/xr/agg ~12.8MB) is L2-resident.
// ---------------------------------------------------------------------------

typedef __attribute__((ext_vector_type(2))) float v2f;
typedef __attribute__((ext_vector_type(8))) float v8f;

static __device__ __forceinline__ float leaky02(float x) {
  return x > 0.f ? x : 0.2f * x;
}

// float atomic max via signed/unsigned int ordering trick (works for mixed signs)
static __device__ __forceinline__ void atomicMaxF32(float* addr, float val) {
  if (val >= 0.f) atomicMax((int*)addr, __float_as_int(val));
  else            atomicMin((unsigned int*)addr, __float_as_uint(val));
}

// ---------------------------------------------------------------------------
// Y[M x Ncols] = X[M x K] @ W[K x Ncols] + bias   (M%16==0, K%4==0, Ncols%64==0)
// Block = 256 threads (8 waves). W[:, col0:col0+64] staged transposed in LDS
// ([col][K], so B fragments are K-contiguous -> ds_load_b64). One wave computes
// a 16-row x 64-col strip (4 WMMA tiles sharing the A fragment).
// A frag (16x4 f32): lane m = lane&15, elem j = X[m][k + (lane>>4)*2 + j]
// B frag (4x16 f32): lane n = lane&15, elem j = W[k + (lane>>4)*2 + j][n]
// C/D (16x16 f32):   elem v -> row v + (lane>>4)*8, col lane&15
// ---------------------------------------------------------------------------
__global__ __launch_bounds__(256)
void gemm_bias_wmma(const float* __restrict__ X, const float* __restrict__ W,
                    const float* __restrict__ bias, float* __restrict__ Y,
                    int K, int Ncols, int n_row_strips) {
  __shared__ float wlds[64 * 128];                // transposed: wlds[c*K + r]
  const int tid  = threadIdx.x;
  const int col0 = blockIdx.y << 6;

  for (int i = tid; i < (K << 6); i += 256) {     // cooperative transpose load
    int c = i / K, r = i - c * K;
    wlds[i] = W[(size_t)r * Ncols + col0 + c];
  }
  __syncthreads();

  const int wave = tid >> 5;
  const int lane = tid & 31;
  const int rs   = blockIdx.x * 8 + wave;
  if (rs >= n_row_strips) return;                 // wave-uniform, after barrier

  const int row0 = rs << 4;
  const int m    = lane & 15;
  const int klo  = (lane >> 4) << 1;              // 0 or 2

  v8f c0 = {}, c1 = {}, c2 = {}, c3 = {};
  const float* xrow = X + (size_t)(row0 + m) * K;
  const float* bp0 = wlds + ( 0 + m) * K + klo;
  const float* bp1 = wlds + (16 + m) * K + klo;
  const float* bp2 = wlds + (32 + m) * K + klo;
  const float* bp3 = wlds + (48 + m) * K + klo;

  for (int k = 0; k < K; k += 4) {
    v2f a  = *(const v2f*)(xrow + k + klo);       // 8B aligned
    v2f b0 = *(const v2f*)(bp0 + k);
    v2f b1 = *(const v2f*)(bp1 + k);
    v2f b2 = *(const v2f*)(bp2 + k);
    v2f b3 = *(const v2f*)(bp3 + k);
    c0 = __builtin_amdgcn_wmma_f32_16x16x4_f32(false, a, false, b0, (short)0, c0, false, false);
    c1 = __builtin_amdgcn_wmma_f32_16x16x4_f32(false, a, false, b1, (short)0, c1, false, false);
    c2 = __builtin_amdgcn_wmma_f32_16x16x4_f32(false, a, false, b2, (short)0, c2, false, false);
    c3 = __builtin_amdgcn_wmma_f32_16x16x4_f32(false, a, false, b3, (short)0, c3, false, false);
  }

  const float bv0 = bias[col0 + m];
  const float bv1 = bias[col0 + 16 + m];
  const float bv2 = bias[col0 + 32 + m];
  const float bv3 = bias[col0 + 48 + m];
  const int rbase = row0 + ((lane >> 4) << 3);
#pragma unroll
  for (int v = 0; v < 8; ++v) {
    size_t base = (size_t)(rbase + v) * Ncols + col0 + m;
    Y[base +  0] = c0[v] + bv0;
    Y[base + 16] = c1[v] + bv1;
    Y[base + 32] = c2[v] + bv2;
    Y[base + 48] = c3[v] + bv3;
  }
}

// ---------------------------------------------------------------------------
__global__ void fill_kernel(float* __restrict__ p, float v, int n) {
  int i = blockIdx.x * blockDim.x + threadIdx.x;
  if (i < n) p[i] = v;
}

// pass 1: logits[e,h] = sum_c leaky(xl[src][h,c]+xr[dst][h,c])*att[h,c]; atomicMax per dst
__global__ void edge_logits_max(const float* __restrict__ xl, const float* __restrict__ xr,
                                const float* __restrict__ att, const int* __restrict__ ei,
                                float* __restrict__ logits, float* __restrict__ maxb,
                                int E, int Etot, int C) {
  int t = blockIdx.x * blockDim.x + threadIdx.x;
  if (t >= (Etot << 3)) return;
  int e = t >> 3, hd = t & 7;                     // H = 8
  int s, d;
  if (e < E) { s = ei[e]; d = ei[E + e]; } else { s = d = e - E; }
  const int HC = C << 3;
  const float* pl = xl + (size_t)s * HC + hd * C;
  const float* pr = xr + (size_t)d * HC + hd * C;
  const float* pa = att + hd * C;
  float acc = 0.f;
  for (int c = 0; c < C; ++c) acc += leaky02(pl[c] + pr[c]) * pa[c];
  logits[t] = acc;
  atomicMaxF32(&maxb[(d << 3) + hd], acc);
}

// pass 2 (fused): a = exp(logit - max[dst]); denom[dst] += a; agg[dst] += a*xl[src]
__global__ void edge_softmax_scatter(const float* __restrict__ xl,
                                     const float* __restrict__ maxb,
                                     const int* __restrict__ ei,
                                     const float* __restrict__ logits,
                                     float* __restrict__ denom, float* __restrict__ agg,
                                     int E, int Etot, int C) {
  int t = blockIdx.x * blockDim.x + threadIdx.x;
  if (t >= (Etot << 3)) return;
  int e = t >> 3, hd = t & 7;                     // H = 8
  int s, d;
  if (e < E) { s = ei[e]; d = ei[E + e]; } else { s = d = e - E; }
  float a = expf(logits[t] - maxb[(d << 3) + hd]);
  atomicAdd(&denom[(d << 3) + hd], a);
  const int HC = C << 3;
  const float* pl = xl + (size_t)s * HC + hd * C;
  float* pd = agg + (size_t)d * HC + hd * C;
  for (int c = 0; c < C; ++c) atomicAdd(&pd[c], a * pl[c]);
}

// h_out = leaky( agg/(denom+eps) + bo + res )   (layers 1..4, HC = 64, C = 8)
__global__ void finalize_layer(const float* __restrict__ agg, const float* __restrict__ denom,
                               const float* __restrict__ res, const float* __restrict__ bo,
                               float* __restrict__ hout, int N) {
  int t = blockIdx.x * blockDim.x + threadIdx.x;
  if (t >= (N << 6)) return;
  int n  = t >> 6;
  int j  = t & 63;
  int hd = j >> 3;
  float v = agg[t] / (denom[(n << 3) + hd] + 1e-16f) + bo[j] + res[t];
  hout[t] = leaky02(v);
}

// layer 5 part 1: z[n,cls] = mean_h( agg[n,h,cls]/(denom[n,h]+eps) ) + bo5[cls]
__global__ void head_mean(const float* __restrict__ agg, const float* __restrict__ denom,
                          const float* __restrict__ bo5, float* __restrict__ z, int N) {
  int t = blockIdx.x * blockDim.x + threadIdx.x;
  if (t >= N * 40) return;
  int n = t / 40, cls = t - n * 40;
  float s = 0.f;
#pragma unroll
  for (int hd = 0; hd < 8; ++hd)
    s += agg[(size_t)n * 320 + hd * 40 + cls] / (denom[(n << 3) + hd] + 1e-16f);
  z[t] = s * 0.125f + bo5[cls];
}

// layer 5 part 2: one wave per node, log_softmax over 40 classes via shuffles
__global__ void logsoftmax40(const float* __restrict__ z, float* __restrict__ out, int N) {
  int gw   = (blockIdx.x * blockDim.x + threadIdx.x) >> 5;  // wave index = node
  int lane = threadIdx.x & 31;
  if (gw >= N) return;                                      // wave-uniform
  const float* zp = z + (size_t)gw * 40;
  float v0 = zp[lane];
  float v1 = (lane < 8) ? zp[32 + lane] : -INFINITY;
  float mx = fmaxf(v0, v1);
#pragma unroll
  for (int off = 16; off; off >>= 1) mx = fmaxf(mx, __shfl_xor(mx, off, 32));
  float se = expf(v0 - mx) + ((lane < 8) ? expf(v1 - mx) : 0.f);
#pragma unroll
  for (int off = 16; off; off >>= 1) se += __shfl_xor(se, off, 32);
  float lse = mx + logf(se);
  float* op = out + (size_t)gw * 40;
  op[lane] = v0 - lse;
  if (lane < 8) op[32 + lane] = v1 - lse;
}

// ---------------------------------------------------------------------------
extern "C" void kernel_launch(void* const* d_in, const int* in_sizes, int n_in,
                              void* d_out, int out_size, void* d_ws, size_t ws_size,
                              hipStream_t stream) {
  const int F = 128, H = 8;
  const int N = in_sizes[0] / F;       // 50000
  const int E = in_sizes[1] / 2;       // 400000
  const int Etot = E + N;              // +self loops
  const float* x  = (const float*)d_in[0];
  const int*   ei = (const int*)d_in[1];

  const float *Wl[5], *bl[5], *Wr[5], *br[5], *att[5], *bo[5];
  for (int i = 0; i < 5; ++i) {
    Wl[i]  = (const float*)d_in[2 + 6 * i + 0];
    bl[i]  = (const float*)d_in[2 + 6 * i + 1];
    Wr[i]  = (const float*)d_in[2 + 6 * i + 2];
    br[i]  = (const float*)d_in[2 + 6 * i + 3];
    att[i] = (const float*)d_in[2 + 6 * i + 4];
    bo[i]  = (const float*)d_in[2 + 6 * i + 5];
  }
  const float *Rw[4], *Rb[4];
  for (int i = 0; i < 4; ++i) {
    Rw[i] = (const float*)d_in[32 + 2 * i];
    Rb[i] = (const float*)d_in[33 + 2 * i];
  }

  // --- workspace carving (256B aligned) ---
  size_t off = 0;
  auto take = [&](size_t bytes) -> float* {
    float* p = (float*)((char*)d_ws + off);
    off += (bytes + 255) & ~(size_t)255;
    return p;
  };
  float* hA  = take((size_t)N * 64 * 4);
  float* hB  = take((size_t)N * 64 * 4);
  float* xl  = take((size_t)N * 320 * 4);   // sized for layer 5
  float* xr  = take((size_t)N * 320 * 4);
  float* res = take((size_t)N * 64 * 4);    // also reused as z[N,40] in layer 5
  float* lg  = take((size_t)Etot * H * 4);
  float* mxb = take((size_t)N * H * 4);
  float* dnm = take((size_t)N * H * 4);
  float* agg = take((size_t)N * 320 * 4);

  const int nrs = N / 16;                    // 3125 row strips (N % 16 == 0)
  auto gemm = [&](const float* X, const float* W, const float* B, float* Y,
                  int K, int Ncols) {
    dim3 g((nrs + 7) / 8, Ncols / 64);
    gemm_bias_wmma<<<g, 256, 0, stream>>>(X, W, B, Y, K, Ncols, nrs);
  };
  auto fill = [&](float* p, float v, int n) {
    fill_kernel<<<(n + 255) / 256, 256, 0, stream>>>(p, v, n);
  };

  const int ETH = Etot * H;
  const int egrid = (ETH + 255) / 256;

  const float* hin = x;
  int K = F;
  float* hbuf[2] = { hA, hB };

  for (int i = 0; i < 4; ++i) {
    gemm(hin, Wl[i], bl[i], xl, K, 64);
    gemm(hin, Wr[i], br[i], xr, K, 64);
    gemm(hin, Rw[i], Rb[i], res, K, 64);
    fill(mxb, -INFINITY, N * H);
    fill(dnm, 0.f, N * H);
    fill(agg, 0.f, N * 64);
    edge_logits_max<<<egrid, 256, 0, stream>>>(xl, xr, att[i], ei, lg, mxb, E, Etot, 8);
    edge_softmax_scatter<<<egrid, 256, 0, stream>>>(xl, mxb, ei, lg, dnm, agg, E, Etot, 8);
    int nt = N * 64;
    finalize_layer<<<(nt + 255) / 256, 256, 0, stream>>>(agg, dnm, res, bo[i],
                                                         hbuf[i & 1], N);
    hin = hbuf[i & 1];
    K = 64;
  }

  // layer 5: H=8, C=40, concat=false (mean over heads) + log_softmax
  gemm(hin, Wl[4], bl[4], xl, 64, 320);
  gemm(hin, Wr[4], br[4], xr, 64, 320);
  fill(mxb, -INFINITY, N * H);
  fill(dnm, 0.f, N * H);
  fill(agg, 0.f, N * 320);
  edge_logits_max<<<egrid, 256, 0, stream>>>(xl, xr, att[4], ei, lg, mxb, E, Etot, 40);
  edge_softmax_scatter<<<egrid, 256, 0, stream>>>(xl, mxb, ei, lg, dnm, agg, E, Etot, 40);
  float* z = res;                            // reuse: N*40 <= N*64
  head_mean<<<(N * 40 + 255) / 256, 256, 0, stream>>>(agg, dnm, bo[4], z, N);
  logsoftmax40<<<(N * 32 + 255) / 256, 256, 0, stream>>>(z, (float*)d_out, N);
}